// LengthRegulator_23940147708156
// MI455X (gfx1250) — compile-verified
//
#include <hip/hip_runtime.h>
#include <hip/hip_bf16.h>
#include <stdint.h>

// Problem constants from the reference
#define LR_T      256
#define LR_D      384
#define LR_MAXLEN 2048
#define WAVES_PER_BLOCK 8   // 256 threads = 8 wave32s
#define LANES     32        // wave32 (CDNA5)

typedef float v4f __attribute__((ext_vector_type(4)));

// ---------------------------------------------------------------------------
// Kernel A: per-batch inclusive scan of duration -> csum (workspace),
//           plus mel_len = min(csum[T-1], max_len) into d_out's int32 tail.
// grid = B blocks, block = 256 threads = 8 wave32s.
// Wave-level shuffle scan (5 steps) + 8-entry LDS cross-wave combine.
// ---------------------------------------------------------------------------
__global__ __launch_bounds__(LR_T)
void lr_scan(const int* __restrict__ dur,
             const int* __restrict__ max_len_p,
             int* __restrict__ csum_out,
             int* __restrict__ mel_out) {
    __shared__ int s_wsum[WAVES_PER_BLOCK];

    const int b    = blockIdx.x;
    const int tid  = threadIdx.x;
    const int wave = tid >> 5;
    const int lane = tid & (LANES - 1);

    int scan = dur[b * LR_T + tid];

    // intra-wave inclusive scan (wave32: 5 rounds of shfl_up)
#pragma unroll
    for (int off = 1; off < LANES; off <<= 1) {
        int n = __shfl_up(scan, off, LANES);
        if (lane >= off) scan += n;
    }

    if (lane == LANES - 1) s_wsum[wave] = scan;   // wave totals
    __syncthreads();

    int base = 0, total = 0;
#pragma unroll
    for (int w = 0; w < WAVES_PER_BLOCK; ++w) {
        const int ws = s_wsum[w];
        if (w < wave) base += ws;
        total += ws;
    }

    csum_out[b * LR_T + tid] = scan + base;

    if (tid == LR_T - 1) {
        const int ml = *max_len_p;
        mel_out[b] = total < ml ? total : ml;
    }
}

// ---------------------------------------------------------------------------
// Kernel B: gather. grid = (MAXLEN/8, B), block = 256 threads (8 waves).
// Stage csum[b][0..255] into LDS with the CDNA5 async global->LDS copy
// (ASYNCcnt-tracked), then each wave binary-searches one output frame and
// streams its 1536-byte row with non-temporal b128 stores (output is
// write-once; NT keeps x resident in the 192MB L2 for the ~8x row re-reads).
// ---------------------------------------------------------------------------
__global__ __launch_bounds__(LR_T)
void lr_gather(const float* __restrict__ x,
               const int*   __restrict__ csum,
               const int*   __restrict__ max_len_p,
               float*       __restrict__ out) {
    __shared__ int s_csum[LR_T];

    const int b   = blockIdx.y;
    const int tid = threadIdx.x;

    // ---- CDNA5 async path: global -> LDS, tracked by ASYNCcnt ----
    {
        unsigned lds_addr = (unsigned)(uintptr_t)&s_csum[tid];
        unsigned long long gaddr =
            (unsigned long long)(uintptr_t)(csum + (size_t)b * LR_T + tid);
        asm volatile("global_load_async_to_lds_b32 %0, %1, off"
                     :: "v"(lds_addr), "v"(gaddr)
                     : "memory");
        asm volatile("s_wait_asynccnt 0" ::: "memory");
    }
    __syncthreads();  // make LDS csum visible to all 8 waves

    const int max_len = *max_len_p;           // uniform scalar load
    const int total   = s_csum[LR_T - 1];
    const int mel_len = total < max_len ? total : max_len;

    const int wave = tid >> 5;
    const int lane = tid & (LANES - 1);
    const int p    = blockIdx.x * WAVES_PER_BLOCK + wave;   // output frame

    // searchsorted(csum, p, side='right') == #(csum <= p); branch-free
    // 8-step binary search; mid is wave-uniform -> LDS broadcast reads.
    int lo = 0, hi = LR_T;
#pragma unroll
    for (int it = 0; it < 8; ++it) {
        const int  mid = (lo + hi) >> 1;
        const bool le  = (s_csum[mid] <= p);
        lo = le ? (mid + 1) : lo;
        hi = le ? hi : mid;
    }
    int idx = lo;
    if (idx > LR_T - 1) idx = LR_T - 1;

    const bool valid = (p < mel_len);

    const v4f* __restrict__ src =
        (const v4f*)(x + ((size_t)b * LR_T + idx) * LR_D);
    v4f* __restrict__ dst =
        (v4f*)(out + ((size_t)b * LR_MAXLEN + p) * LR_D);

    const v4f z = (v4f)(0.0f);
#pragma unroll
    for (int c = 0; c < 3; ++c) {
        const int off = lane + LANES * c;     // 0..95: wave covers whole row
        v4f v = z;
        if (valid) v = src[off];              // global_load_b128 (coalesced, cached)
        __builtin_nontemporal_store(v, &dst[off]);  // b128 store th:NT (streaming)
    }
}

// ---------------------------------------------------------------------------
extern "C" void kernel_launch(void* const* d_in, const int* in_sizes, int n_in,
                              void* d_out, int out_size, void* d_ws, size_t ws_size,
                              hipStream_t stream) {
    const float* x       = (const float*)d_in[0];   // [B, T, D] f32
    const int*   dur     = (const int*)d_in[1];     // [B, T] i32
    const int*   max_len = (const int*)d_in[2];     // scalar i32

    const int B = in_sizes[1] / LR_T;               // 32

    float* out = (float*)d_out;                                   // [B, MAXLEN, D] f32
    int*   mel = (int*)(out + (size_t)B * LR_MAXLEN * LR_D);      // [B] i32 tail
    int*   csum_ws = (int*)d_ws;                                  // [B, T] i32 scratch

    lr_scan<<<dim3(B), dim3(LR_T), 0, stream>>>(dur, max_len, csum_ws, mel);

    dim3 grid(LR_MAXLEN / WAVES_PER_BLOCK, B);      // (256, 32)
    lr_gather<<<grid, dim3(LR_T), 0, stream>>>(x, csum_ws, max_len, out);
}